// TemporalGNN_46514495815770
// MI455X (gfx1250) — compile-verified
//
#include <hip/hip_runtime.h>

// ---------------------------------------------------------------------------
// GraphSAGE 2-layer GNN for MI455X (gfx1250, wave32)
//   layer(h) = relu?( (mean-agg h) @ W_l^T + b_l + h @ W_r^T )
// Scatter phase: L2-resident random traffic (node table = 25.6MB << 192MB L2).
// GEMM phase: exact-precision V_WMMA_F32_16X16X4_F32 with TWO interleaved
// independent accumulator chains so A-fragment loads overlap WMMA execution
// instead of serializing behind a single D->C dependency chain.
// ---------------------------------------------------------------------------

typedef float v2f __attribute__((ext_vector_type(2)));
typedef float v8f __attribute__((ext_vector_type(8)));

#if defined(__gfx1250__) && __has_builtin(__builtin_amdgcn_wmma_f32_16x16x4_f32)
#define USE_WMMA_F32 1
#else
#define USE_WMMA_F32 0
#endif

#define FEAT 64  // F_IN == HID == OUT == 64

// ---------------------------------------------------------------------------
__global__ __launch_bounds__(256) void fill_zero_kernel(float* __restrict__ p,
                                                        long long n) {
  long long i = (long long)blockIdx.x * blockDim.x + threadIdx.x;
  long long stride = (long long)gridDim.x * blockDim.x;
  for (; i < n; i += stride) p[i] = 0.0f;
}

// One edge per 16 lanes; lane does a coalesced float4 gather of the src row
// and 4 f32 atomic adds into the dst row. Node tables are L2-resident.
__global__ __launch_bounds__(256) void scatter_kernel(
    const float* __restrict__ h, const long long* __restrict__ ei,
    float* __restrict__ sum, float* __restrict__ cnt, int E, int addCnt) {
  int gid = blockIdx.x * 256 + threadIdx.x;
  int edge = gid >> 4;  // 16 threads per edge
  int part = gid & 15;  // each covers 4 consecutive floats
  if (edge >= E) return;
  long long s = ei[edge];                 // edge_index[0][e] = src
  long long d = ei[(long long)E + edge];  // edge_index[1][e] = dst
  float4 v = ((const float4*)(h + s * FEAT))[part];
  float* dr = sum + d * FEAT + part * 4;
  atomicAdd(dr + 0, v.x);
  atomicAdd(dr + 1, v.y);
  atomicAdd(dr + 2, v.z);
  atomicAdd(dr + 3, v.w);
  if (addCnt && part == 0) atomicAdd(cnt + d, 1.0f);
}

// cnt[i] := 1 / max(cnt[i], 1)   (degree is identical for both layers)
__global__ __launch_bounds__(256) void invcnt_kernel(float* __restrict__ cnt,
                                                     int n) {
  int i = blockIdx.x * 256 + threadIdx.x;
  if (i < n) cnt[i] = 1.0f / fmaxf(cnt[i], 1.0f);
}

// ---------------------------------------------------------------------------
// out[i,:] = relu?( (sum[i,:]*invc[i]) @ Wl^T + bl + h[i,:] @ Wr^T )
// 256 threads = 8 waves; block covers 64 nodes = 4 row-tiles x 4 col-tiles of
// 16x16; each wave computes 2 tiles with dual interleaved WMMA chains.
__global__ __launch_bounds__(256, 1) void sage_layer_kernel(
    const float* __restrict__ h, const float* __restrict__ sum,
    const float* __restrict__ invc, const float* __restrict__ Wl,
    const float* __restrict__ bl, const float* __restrict__ Wr,
    float* __restrict__ out, int n, int relu) {
  // Stride-65 padding: B-fragment column reads hit distinct LDS banks.
  __shared__ float sWl[FEAT * 65];
  __shared__ float sWr[FEAT * 65];
  __shared__ float sB[FEAT];

  int tid = threadIdx.x;
#pragma unroll
  for (int i = 0; i < 16; ++i) {  // 4096 elements / 256 threads
    int e = tid + 256 * i;
    int r = e >> 6, c = e & 63;
    sWl[r * 65 + c] = Wl[e];
    sWr[r * 65 + c] = Wr[e];
  }
  if (tid < FEAT) sB[tid] = bl[tid];
  __syncthreads();

  int wave = tid >> 5;
  int lane = tid & 31;
  int lhalf = lane & 15;  // lane within half-wave
  int khi = lane >> 4;    // 0: K%4 in {0,1}; 1: K%4 in {2,3}  (A/B f32 layout)
  int base = blockIdx.x * 64;

  for (int t = wave; t < 16; t += 8) {
    int rbase = base + (t >> 2) * 16;  // node row of this 16x16 tile
    int col0 = (t & 3) * 16;           // output column of this tile

    // A-fragment row (clamped so EXEC stays all-1s for WMMA; store predicated)
    int arow = rbase + lhalf;
    if (arow > n - 1) arow = n - 1;
    const float* aggRow = sum + (size_t)arow * FEAT;
    const float* hRow = h + (size_t)arow * FEAT;
    float ic = invc[arow];
    const float* wlRow = &sWl[(col0 + lhalf) * 65];  // B[k][n] = Wl[n][k]
    const float* wrRow = &sWr[(col0 + lhalf) * 65];

    v8f accA = {0.f, 0.f, 0.f, 0.f, 0.f, 0.f, 0.f, 0.f};
    v8f accH = {0.f, 0.f, 0.f, 0.f, 0.f, 0.f, 0.f, 0.f};

#if USE_WMMA_F32
    // Two INDEPENDENT accumulator chains, interleaved: while one chain's
    // A-fragment load is outstanding, the other chain's WMMA executes.
#pragma unroll
    for (int kk = 0; kk < 16; ++kk) {
      int k0 = kk * 4 + khi * 2;  // 8B-aligned (k0 even, row base 256B-aligned)
      v2f a = *(const v2f*)(aggRow + k0);  // agg fragment (global, L2-hot)
      v2f hv = *(const v2f*)(hRow + k0);   // h   fragment (global, L2-hot)
      v2f as, bL, bR;
      as.x = a.x * ic;  // fused mean-normalization
      as.y = a.y * ic;
      bL.x = wlRow[k0];      // B VGPR0: row K=k0,   col = col0+lhalf
      bL.y = wlRow[k0 + 1];  // B VGPR1: row K=k0+1
      bR.x = wrRow[k0];
      bR.y = wrRow[k0 + 1];
      accA = __builtin_amdgcn_wmma_f32_16x16x4_f32(false, as, false, bL,
                                                   (short)0, accA, false, false);
      accH = __builtin_amdgcn_wmma_f32_16x16x4_f32(false, hv, false, bR,
                                                   (short)0, accH, false, false);
    }
#else
    (void)aggRow; (void)hRow; (void)ic;
    // Scalar fallback (same output-lane mapping as the WMMA C/D layout).
#pragma unroll
    for (int r = 0; r < 8; ++r) {
      int m = rbase + r + 8 * khi;
      if (m > n - 1) m = n - 1;
      float icm = invc[m];
      const float* ar = sum + (size_t)m * FEAT;
      const float* hr = h + (size_t)m * FEAT;
      float a0 = 0.f, h0 = 0.f;
      for (int k = 0; k < FEAT; ++k) {
        a0 += ar[k] * icm * wlRow[k];
        h0 += hr[k] * wrRow[k];
      }
      accA[r] = a0;
      accH[r] = h0;
    }
#endif

    // C/D layout: VGPR r -> M = r + 8*khi, N = lhalf
    float bb = sB[col0 + lhalf];
#pragma unroll
    for (int r = 0; r < 8; ++r) {
      int m = rbase + r + 8 * khi;
      float v = accA[r] + accH[r] + bb;
      if (relu) v = fmaxf(v, 0.f);
      if (m < n) out[(size_t)m * FEAT + col0 + lhalf] = v;
    }
  }
}

// ---------------------------------------------------------------------------
extern "C" void kernel_launch(void* const* d_in, const int* in_sizes, int n_in,
                              void* d_out, int out_size, void* d_ws,
                              size_t ws_size, hipStream_t stream) {
  const float* x = (const float*)d_in[0];
  const long long* ei = (const long long*)d_in[1];  // int64 [2, E]
  const float* W1l = (const float*)d_in[2];
  const float* b1l = (const float*)d_in[3];
  const float* W1r = (const float*)d_in[4];
  const float* W2l = (const float*)d_in[5];
  const float* b2l = (const float*)d_in[6];
  const float* W2r = (const float*)d_in[7];
  float* out = (float*)d_out;

  int n = in_sizes[0] / FEAT;  // 100000
  int E = in_sizes[1] / 2;     // 1000000

  // Workspace: sum [n*64] | cnt/invcnt [n] | hidden h [n*64]  (~51.6 MB)
  float* d_sum = (float*)d_ws;
  float* d_cnt = d_sum + (size_t)n * FEAT;
  float* d_h = d_cnt + n;  // byte offset n*65*4, 16B-aligned for float4 loads

  int sgrid = (E * 16 + 255) / 256;
  int lgrid = (n + 63) / 64;

  // ---- Layer 1: agg(x) -> relu(lin) -> d_h -----------------------------
  fill_zero_kernel<<<2048, 256, 0, stream>>>(d_sum, (long long)n * 65);  // sum+cnt
  scatter_kernel<<<sgrid, 256, 0, stream>>>(x, ei, d_sum, d_cnt, E, 1);
  invcnt_kernel<<<(n + 255) / 256, 256, 0, stream>>>(d_cnt, n);
  sage_layer_kernel<<<lgrid, 256, 0, stream>>>(x, d_sum, d_cnt, W1l, b1l, W1r,
                                               d_h, n, 1);

  // ---- Layer 2: agg(h) -> lin -> d_out ---------------------------------
  fill_zero_kernel<<<2048, 256, 0, stream>>>(d_sum, (long long)n * FEAT);
  scatter_kernel<<<sgrid, 256, 0, stream>>>(d_h, ei, d_sum, d_cnt, E, 0);
  sage_layer_kernel<<<lgrid, 256, 0, stream>>>(d_h, d_sum, d_cnt, W2l, b2l,
                                               W2r, out, n, 0);
}